// MinCutPooledConvolutionalNetwork_79800492359761
// MI455X (gfx1250) — compile-verified
//
#include <hip/hip_runtime.h>
#include <hip/hip_bf16.h>
#include <math.h>

// Problem constants (from reference)
#define Bg   128
#define Nn   512
#define INC  128
#define HID  32
#define K1C  100
#define K2C  10
#define OUTC 10

typedef float v2f __attribute__((ext_vector_type(2)));
typedef float v8f __attribute__((ext_vector_type(8)));

// D = A(16x4) x B(4x16) + C(16x16), fp32, wave32.
// A layout: lanes 0-15 rows M=0..15 holding K=(0,1); lanes 16-31 same rows holding K=(2,3).
// B layout: lanes 0-15 cols N=0..15 holding K=(0,1); lanes 16-31 same cols holding K=(2,3).
// C layout: VGPR i -> M=i (lanes 0-15) / M=8+i (lanes 16-31), N=lane&15.
__device__ __forceinline__ v8f wmma4(v2f a, v2f b, v8f c) {
  return __builtin_amdgcn_wmma_f32_16x16x4_f32(false, a, false, b, (short)0, c, false, false);
}

// CDNA5 async global->LDS copy (ISA 10.x / 15.18.3): per-lane 16B, tracked by ASYNCcnt.
// lds_off = LDS byte address (low 32 bits of generic pointer per aperture mapping).
__device__ __forceinline__ void async_g2l_b128(uint32_t lds_off, const void* g) {
  asm volatile("global_load_async_to_lds_b128 %0, %1, off"
               :: "v"(lds_off), "v"((unsigned long long)(uintptr_t)g)
               : "memory");
}
__device__ __forceinline__ void wait_async0() {
  asm volatile("s_wait_asynccnt 0x0" ::: "memory");
}
__device__ __forceinline__ uint32_t lds_addr(const void* p) {
  return (uint32_t)(uintptr_t)p;
}

// ---------------------------------------------------------------- adjacency
__global__ void k_build_adj(const int* __restrict__ ei, int E,
                            float* __restrict__ adj, float* __restrict__ degin,
                            float* __restrict__ rowsum) {
  int e = blockIdx.x * blockDim.x + threadIdx.x;
  if (e >= E) return;
  int src = ei[e];
  int dst = ei[E + e];
  int b  = src >> 9;       // / 512
  int sl = src & 511;
  int dl = dst & 511;
  atomicAdd(&adj[((size_t)b * Nn + sl) * Nn + dl], 1.0f);
  atomicAdd(&degin[dst], 1.0f);    // GCN in-degree (self loop added at use)
  atomicAdd(&rowsum[src], 1.0f);   // d_flat = adj.sum(-1)
}

// ---------------------------------------------------------------- xw = x @ W1 (WMMA)
__global__ void k_xw(const float* __restrict__ x, const float* __restrict__ W,
                     float* __restrict__ xw) {
  const int tm = blockIdx.x * 16, tn = blockIdx.y * 16;
  const int lane = threadIdx.x, m = lane & 15, hh = lane >> 4;
  v8f c = {};
  for (int k0 = 0; k0 < INC; k0 += 4) {
    const int kb = k0 + 2 * hh;
    v2f a = { x[(size_t)(tm + m) * INC + kb], x[(size_t)(tm + m) * INC + kb + 1] };
    v2f b = { W[(size_t)kb * HID + tn + m],  W[(size_t)(kb + 1) * HID + tn + m] };
    c = wmma4(a, b, c);
  }
  for (int i = 0; i < 8; ++i)
    xw[(size_t)(tm + 8 * hh + i) * HID + tn + m] = c[i];
}

// ------------------------------------------ GCN aggregation: h = relu(Dis*(A^T (Dis*xw)) + Dis^2*xw + b)
// One workgroup (32 waves) per graph. xw slab (64KB) + degrees staged in LDS via async
// copies; each wave owns one 16-row M-strip and fuses both 16-wide N-tiles, so the
// 1MB per-graph adjacency is streamed exactly once.
__global__ __launch_bounds__(1024) void k_gcn(const float* __restrict__ adj,
                                              const float* __restrict__ xw,
                                              const float* __restrict__ degin,
                                              const float* __restrict__ bias,
                                              float* __restrict__ hout) {
  extern __shared__ float smem_g[];
  float* xls = smem_g;             // Nn*HID = 16384 floats
  float* dls = smem_g + Nn * HID;  // Nn floats
  const int b   = blockIdx.x;
  const int tid = threadIdx.x;

  {
    const float* xsrc = xw + (size_t)b * Nn * HID;
    const uint32_t xbase = lds_addr(xls);
    for (int i = tid; i < (Nn * HID) / 4; i += 1024)
      async_g2l_b128(xbase + (uint32_t)i * 16u, xsrc + (size_t)i * 4);
    const float* dsrc = degin + (size_t)b * Nn;
    const uint32_t dbase = lds_addr(dls);
    if (tid < Nn / 4)
      async_g2l_b128(dbase + (uint32_t)tid * 16u, dsrc + (size_t)tid * 4);
    wait_async0();
  }
  __syncthreads();
  if (tid < Nn) dls[tid] = rsqrtf(dls[tid] + 1.0f);   // dis = rsqrt(deg_in + selfloop)
  __syncthreads();

  const int wave = tid >> 5;
  const int lane = tid & 31;
  const int m = lane & 15, hh = lane >> 4;
  const int tm = wave * 16;
  const float* A = adj + (size_t)b * Nn * Nn;

  v8f c0 = {}, c1 = {};
  for (int k0 = 0; k0 < Nn; k0 += 4) {
    const int kb = k0 + 2 * hh;
    if ((k0 & 63) == 0 && k0 + 64 < Nn)
      __builtin_prefetch(&A[(size_t)(kb + 64) * Nn + tm + m], 0, 1);
    // A'[m][k] = A[k][m]  (coalesced along m)
    v2f a  = { A[(size_t)kb * Nn + tm + m], A[(size_t)(kb + 1) * Nn + tm + m] };
    const float d0 = dls[kb], d1 = dls[kb + 1];
    v2f b0 = { d0 * xls[kb * HID + m],      d1 * xls[(kb + 1) * HID + m] };
    v2f b1 = { d0 * xls[kb * HID + 16 + m], d1 * xls[(kb + 1) * HID + 16 + m] };
    c0 = wmma4(a, b0, c0);
    c1 = wmma4(a, b1, c1);
  }
  for (int i = 0; i < 8; ++i) {
    const int r = tm + 8 * hh + i;
    const float dis = dls[r];
    const size_t node = (size_t)b * Nn + r;
    float v0 = dis * c0[i] + dis * dis * xls[r * HID + m] + bias[m];
    float v1 = dis * c1[i] + dis * dis * xls[r * HID + 16 + m] + bias[16 + m];
    hout[node * HID + m]      = fmaxf(v0, 0.0f);
    hout[node * HID + 16 + m] = fmaxf(v1, 0.0f);
  }
}

// ---------------------------------------------------------------- x1 = s^T h (WMMA, M=100)
__global__ void k_pool_x(const float* __restrict__ s, const float* __restrict__ hf,
                         float* __restrict__ x1) {
  const int b = blockIdx.z;
  const int tm = blockIdx.x * 16, tn = blockIdx.y * 16;
  const int lane = threadIdx.x, m = lane & 15, hh = lane >> 4;
  const int mm = tm + m;
  const bool mv = mm < K1C;
  v8f c = {};
  for (int k0 = 0; k0 < Nn; k0 += 4) {
    const int kb = k0 + 2 * hh;
    v2f a  = { mv ? s[((size_t)b * Nn + kb) * K1C + mm] : 0.0f,
               mv ? s[((size_t)b * Nn + kb + 1) * K1C + mm] : 0.0f };
    v2f bb = { hf[((size_t)b * Nn + kb) * HID + tn + m],
               hf[((size_t)b * Nn + kb + 1) * HID + tn + m] };
    c = wmma4(a, bb, c);
  }
  for (int i = 0; i < 8; ++i) {
    const int row = tm + 8 * hh + i;
    if (row < K1C) x1[((size_t)b * K1C + row) * HID + tn + m] = c[i];
  }
}

// ---------------------------------------------------------------- As = adj @ s (WMMA)
// One workgroup (32 waves) per graph; the full 200KB softmax slab s[512,100] lives in
// LDS (WGP has 320KB); each wave computes a 16-row strip fusing all 7 N-tiles, so the
// per-graph adjacency megabyte is streamed exactly once.
__global__ __launch_bounds__(1024) void k_As(const float* __restrict__ adj,
                                             const float* __restrict__ s,
                                             float* __restrict__ As) {
  extern __shared__ float sl[];                // Nn*K1C = 51200 floats (200KB)
  const int b   = blockIdx.x;
  const int tid = threadIdx.x;
  {
    const float* src = s + (size_t)b * Nn * K1C;
    const uint32_t base = lds_addr(sl);
    for (int i = tid; i < (Nn * K1C) / 4; i += 1024)
      async_g2l_b128(base + (uint32_t)i * 16u, src + (size_t)i * 4);
    wait_async0();
  }
  __syncthreads();

  const int wave = tid >> 5;
  const int lane = tid & 31;
  const int m = lane & 15, hh = lane >> 4;
  const int tm = wave * 16;
  const float* A = adj + ((size_t)b * Nn + tm) * Nn;  // this wave's 16 rows

  v8f c[7];
#pragma unroll
  for (int t7 = 0; t7 < 7; ++t7) c[t7] = v8f{};

  for (int k0 = 0; k0 < Nn; k0 += 4) {
    const int kb = k0 + 2 * hh;
    if ((k0 & 31) == 0 && k0 + 32 < Nn)
      __builtin_prefetch(&A[(size_t)m * Nn + kb + 32], 0, 1);
    v2f a = { A[(size_t)m * Nn + kb], A[(size_t)m * Nn + kb + 1] };
#pragma unroll
    for (int t7 = 0; t7 < 7; ++t7) {
      const int col = t7 * 16 + m;
      const bool cv = col < K1C;
      v2f bb = { cv ? sl[kb * K1C + col] : 0.0f,
                 cv ? sl[(kb + 1) * K1C + col] : 0.0f };
      c[t7] = wmma4(a, bb, c[t7]);
    }
  }
#pragma unroll
  for (int t7 = 0; t7 < 7; ++t7) {
    const int col = t7 * 16 + m;
    if (col < K1C)
      for (int i = 0; i < 8; ++i)
        As[((size_t)b * Nn + tm + 8 * hh + i) * K1C + col] = c[t7][i];
  }
}

// ------------------------------------------------ out = s^T B  (B is [512,100]; used for s^T(As) and s^T s)
__global__ void k_sTB(const float* __restrict__ s, const float* __restrict__ Bm,
                      float* __restrict__ out) {
  const int b = blockIdx.z;
  const int tm = blockIdx.x * 16, tn = blockIdx.y * 16;
  const int lane = threadIdx.x, m = lane & 15, hh = lane >> 4;
  const int mm = tm + m;  const bool mv = mm < K1C;
  const int col = tn + m; const bool cv = col < K1C;
  v8f c = {};
  for (int k0 = 0; k0 < Nn; k0 += 4) {
    const int kb = k0 + 2 * hh;
    v2f a  = { mv ? s[((size_t)b * Nn + kb) * K1C + mm] : 0.0f,
               mv ? s[((size_t)b * Nn + kb + 1) * K1C + mm] : 0.0f };
    v2f bb = { cv ? Bm[((size_t)b * Nn + kb) * K1C + col] : 0.0f,
               cv ? Bm[((size_t)b * Nn + kb + 1) * K1C + col] : 0.0f };
    c = wmma4(a, bb, c);
  }
  for (int i = 0; i < 8; ++i) {
    const int row = tm + 8 * hh + i;
    if (row < K1C && cv) out[((size_t)b * K1C + row) * K1C + col] = c[i];
  }
}

// ---------------------------------------------------------------- small dense ops
__global__ void k_s1lin(const float* __restrict__ h, const float* __restrict__ W,
                        const float* __restrict__ bias, float* __restrict__ s1) {
  size_t idx = (size_t)blockIdx.x * blockDim.x + threadIdx.x;
  if (idx >= (size_t)Bg * Nn * K1C) return;
  int j = (int)(idx % K1C);
  size_t i = idx / K1C;
  float acc = bias[j];
  for (int f = 0; f < HID; ++f) acc += h[i * HID + f] * W[f * K1C + j];
  s1[idx] = acc;
}

__global__ void k_softmax100(float* __restrict__ s) {
  size_t row = blockIdx.x;
  int t = threadIdx.x;
  float* p = s + row * K1C;
  float mx = -INFINITY;
  for (int j = t; j < K1C; j += 32) mx = fmaxf(mx, p[j]);
  for (int o = 16; o; o >>= 1) mx = fmaxf(mx, __shfl_xor(mx, o, 32));
  float sum = 0.0f;
  for (int j = t; j < K1C; j += 32) { float e = expf(p[j] - mx); p[j] = e; sum += e; }
  for (int o = 16; o; o >>= 1) sum += __shfl_xor(sum, o, 32);
  float inv = 1.0f / sum;
  for (int j = t; j < K1C; j += 32) p[j] *= inv;
}

// mincut / ortho losses (generic over node count NN, cluster count KC)
__global__ void k_losses(const float* __restrict__ adjraw, const float* __restrict__ ss,
                         const float* __restrict__ s, const float* __restrict__ dflat,
                         int NN, int KC,
                         float* __restrict__ num, float* __restrict__ den,
                         float* __restrict__ orth) {
  const int b = blockIdx.x, t = threadIdx.x, nt = blockDim.x;
  __shared__ float red[256];
  __shared__ float fro_s;
  float acc = 0.0f;
  for (int k = t; k < KC; k += nt) acc += adjraw[((size_t)b * KC + k) * KC + k];
  red[t] = acc; __syncthreads();
  for (int o = nt >> 1; o; o >>= 1) { if (t < o) red[t] += red[t + o]; __syncthreads(); }
  if (t == 0) num[b] = red[0];
  __syncthreads();
  acc = 0.0f;
  for (int n = t; n < NN; n += nt) {
    const float* sp = s + ((size_t)b * NN + n) * KC;
    float ssum = 0.0f;
    for (int k = 0; k < KC; ++k) ssum += sp[k] * sp[k];
    acc += dflat[b * NN + n] * ssum;
  }
  red[t] = acc; __syncthreads();
  for (int o = nt >> 1; o; o >>= 1) { if (t < o) red[t] += red[t + o]; __syncthreads(); }
  if (t == 0) den[b] = red[0];
  __syncthreads();
  const int KK = KC * KC;
  acc = 0.0f;
  for (int i = t; i < KK; i += nt) { float v = ss[(size_t)b * KK + i]; acc += v * v; }
  red[t] = acc; __syncthreads();
  for (int o = nt >> 1; o; o >>= 1) { if (t < o) red[t] += red[t + o]; __syncthreads(); }
  if (t == 0) fro_s = sqrtf(red[0]);
  __syncthreads();
  const float invf = 1.0f / fro_s;
  const float ik = rsqrtf((float)KC);
  acc = 0.0f;
  for (int i = t; i < KK; i += nt) {
    int r = i / KC, cc = i % KC;
    float v = ss[(size_t)b * KK + i] * invf - ((r == cc) ? ik : 0.0f);
    acc += v * v;
  }
  red[t] = acc; __syncthreads();
  for (int o = nt >> 1; o; o >>= 1) { if (t < o) red[t] += red[t + o]; __syncthreads(); }
  if (t == 0) orth[b] = sqrtf(red[0]);
}

__global__ void k_adjd(const float* __restrict__ adjraw, float* __restrict__ d, int KC) {
  const int b = blockIdx.x, i = threadIdx.x;
  if (i >= KC) return;
  const float* r = adjraw + ((size_t)b * KC + i) * KC;
  float sum = 0.0f;
  for (int j = 0; j < KC; ++j) if (j != i) sum += r[j];
  d[b * KC + i] = sqrtf(sum) + 1e-15f;
}

__global__ void k_adjnorm(float* __restrict__ adj, const float* __restrict__ d, int KC) {
  size_t idx = (size_t)blockIdx.x * blockDim.x + threadIdx.x;
  size_t tot = (size_t)Bg * KC * KC;
  if (idx >= tot) return;
  int j = (int)(idx % KC);
  size_t r = idx / KC;
  int i = (int)(r % KC);
  int b = (int)(r / KC);
  float v = (i == j) ? 0.0f : adj[idx];
  adj[idx] = v / (d[b * KC + i] * d[b * KC + j]);
}

__global__ void k_rowsum(const float* __restrict__ adj, float* __restrict__ out, int KC) {
  size_t idx = (size_t)blockIdx.x * blockDim.x + threadIdx.x;
  if (idx >= (size_t)Bg * KC) return;
  const float* r = adj + idx * KC;
  float sum = 0.0f;
  for (int j = 0; j < KC; ++j) sum += r[j];
  out[idx] = sum;
}

// t[b][i][f] = sum_j adj[b][i][j] * x[b][j][f]
__global__ void k_dmm_adj_x(const float* __restrict__ adj, const float* __restrict__ xin,
                            float* __restrict__ out, int K, int F) {
  size_t idx = (size_t)blockIdx.x * blockDim.x + threadIdx.x;
  size_t tot = (size_t)Bg * K * F;
  if (idx >= tot) return;
  int f = (int)(idx % F);
  size_t r = idx / F;
  int i = (int)(r % K);
  int b = (int)(r / K);
  float acc = 0.0f;
  for (int j = 0; j < K; ++j)
    acc += adj[((size_t)b * K + i) * K + j] * xin[((size_t)b * K + j) * F + f];
  out[idx] = acc;
}

// out = t@relW + relb + x@rootW  (optional relu)
__global__ void k_gconv_fin(const float* __restrict__ t, const float* __restrict__ xin,
                            const float* __restrict__ relW, const float* __restrict__ relb,
                            const float* __restrict__ rootW, float* __restrict__ out,
                            int K, int doRelu) {
  size_t idx = (size_t)blockIdx.x * blockDim.x + threadIdx.x;
  size_t tot = (size_t)Bg * K * HID;
  if (idx >= tot) return;
  int f = (int)(idx % HID);
  size_t node = idx / HID;
  float acc = relb[f];
  for (int g = 0; g < HID; ++g)
    acc += t[node * HID + g] * relW[g * HID + f] + xin[node * HID + g] * rootW[g * HID + f];
  out[idx] = doRelu ? fmaxf(acc, 0.0f) : acc;
}

__global__ void k_s2lin(const float* __restrict__ xd, const float* __restrict__ W,
                        const float* __restrict__ bias, float* __restrict__ s2) {
  size_t idx = (size_t)blockIdx.x * blockDim.x + threadIdx.x;
  if (idx >= (size_t)Bg * K1C * K2C) return;
  int j = (int)(idx % K2C);
  size_t node = idx / K2C;
  float acc = bias[j];
  for (int f = 0; f < HID; ++f) acc += xd[node * HID + f] * W[f * K2C + j];
  s2[idx] = acc;
}

__global__ void k_softmax10(float* __restrict__ s) {
  size_t row = (size_t)blockIdx.x * blockDim.x + threadIdx.x;
  if (row >= (size_t)Bg * K1C) return;
  float* p = s + row * K2C;
  float mx = -INFINITY;
  for (int j = 0; j < K2C; ++j) mx = fmaxf(mx, p[j]);
  float sum = 0.0f;
  for (int j = 0; j < K2C; ++j) { float e = expf(p[j] - mx); p[j] = e; sum += e; }
  float inv = 1.0f / sum;
  for (int j = 0; j < K2C; ++j) p[j] *= inv;
}

// out[b][k][f] = sum_n s[b][n][k] * x[b][n][f]
__global__ void k_pool_small(const float* __restrict__ s, const float* __restrict__ xin,
                             float* __restrict__ out, int NNp, int KCp, int F) {
  size_t idx = (size_t)blockIdx.x * blockDim.x + threadIdx.x;
  size_t tot = (size_t)Bg * KCp * F;
  if (idx >= tot) return;
  int f = (int)(idx % F);
  size_t r = idx / F;
  int k = (int)(r % KCp);
  int b = (int)(r / KCp);
  float acc = 0.0f;
  for (int n = 0; n < NNp; ++n)
    acc += s[((size_t)b * NNp + n) * KCp + k] * xin[((size_t)b * NNp + n) * F + f];
  out[idx] = acc;
}

__global__ void k_mean(const float* __restrict__ xd, float* __restrict__ g) {
  size_t idx = (size_t)blockIdx.x * blockDim.x + threadIdx.x;
  if (idx >= (size_t)Bg * HID) return;
  int f = (int)(idx % HID);
  int b = (int)(idx / HID);
  float acc = 0.0f;
  for (int i = 0; i < K2C; ++i) acc += xd[((size_t)b * K2C + i) * HID + f];
  g[idx] = acc * (1.0f / K2C);
}

__global__ void k_lin1(const float* __restrict__ g, const float* __restrict__ W,
                       const float* __restrict__ bias, float* __restrict__ g1) {
  size_t idx = (size_t)blockIdx.x * blockDim.x + threadIdx.x;
  if (idx >= (size_t)Bg * HID) return;
  int f = (int)(idx % HID);
  int b = (int)(idx / HID);
  float acc = bias[f];
  for (int k = 0; k < HID; ++k) acc += g[b * HID + k] * W[k * HID + f];
  g1[idx] = fmaxf(acc, 0.0f);
}

__global__ void k_head(const float* __restrict__ g1, const float* __restrict__ W,
                       const float* __restrict__ bias, float* __restrict__ out) {
  int b = threadIdx.x;
  if (b >= Bg) return;
  float logits[OUTC];
  float mx = -INFINITY;
  for (int c = 0; c < OUTC; ++c) {
    float acc = bias[c];
    for (int f = 0; f < HID; ++f) acc += g1[b * HID + f] * W[f * OUTC + c];
    logits[c] = acc;
    mx = fmaxf(mx, acc);
  }
  float sum = 0.0f;
  for (int c = 0; c < OUTC; ++c) sum += expf(logits[c] - mx);
  float lse = logf(sum) + mx;
  for (int c = 0; c < OUTC; ++c) out[b * OUTC + c] = logits[c] - lse;
}

__global__ void k_floss(const float* n1, const float* d1, const float* o1,
                        const float* n2, const float* d2, const float* o2,
                        float* __restrict__ out) {
  __shared__ float red[128];
  int t = threadIdx.x;
  red[t] = -n1[t] / d1[t] - n2[t] / d2[t] + o1[t] + o2[t];
  __syncthreads();
  for (int o = 64; o; o >>= 1) { if (t < o) red[t] += red[t + o]; __syncthreads(); }
  if (t == 0) out[Bg * OUTC] = red[0] / (float)Bg;
}

// ---------------------------------------------------------------- driver
extern "C" void kernel_launch(void* const* d_in, const int* in_sizes, int n_in,
                              void* d_out, int out_size, void* d_ws, size_t ws_size,
                              hipStream_t stream) {
  (void)n_in; (void)out_size; (void)ws_size;
  const float* x    = (const float*)d_in[0];
  const int*   ei   = (const int*)d_in[1];
  const float* c1W  = (const float*)d_in[3];
  const float* c1b  = (const float*)d_in[4];
  const float* p1W  = (const float*)d_in[5];
  const float* p1b  = (const float*)d_in[6];
  const float* c2rW = (const float*)d_in[7];
  const float* c2rb = (const float*)d_in[8];
  const float* c2oW = (const float*)d_in[9];
  const float* p2W  = (const float*)d_in[10];
  const float* p2b  = (const float*)d_in[11];
  const float* c3rW = (const float*)d_in[12];
  const float* c3rb = (const float*)d_in[13];
  const float* c3oW = (const float*)d_in[14];
  const float* l1W  = (const float*)d_in[15];
  const float* l1b  = (const float*)d_in[16];
  const float* l2W  = (const float*)d_in[17];
  const float* l2b  = (const float*)d_in[18];
  float* out = (float*)d_out;
  const int E   = in_sizes[1] / 2;
  const int TOT = Bg * Nn;

  float* w = (float*)d_ws;
  size_t off = 0;
  auto alloc = [&](size_t n) { float* p = w + off; off += n; return p; };
  float* adj    = alloc((size_t)Bg * Nn * Nn);   // 134 MB
  float* degin  = alloc(TOT);
  float* rowsum = alloc(TOT);
  float* xw     = alloc((size_t)TOT * HID);
  float* hfeat  = alloc((size_t)TOT * HID);
  float* s1     = alloc((size_t)TOT * K1C);
  float* As1    = alloc((size_t)TOT * K1C);
  float* x1     = alloc((size_t)Bg * K1C * HID);
  float* adj1   = alloc((size_t)Bg * K1C * K1C);
  float* ss1    = alloc((size_t)Bg * K1C * K1C);
  float* num1 = alloc(Bg); float* den1 = alloc(Bg); float* or1 = alloc(Bg);
  float* dv1  = alloc((size_t)Bg * K1C);
  float* dfl1 = alloc((size_t)Bg * K1C);
  float* t2   = alloc((size_t)Bg * K1C * HID);
  float* xd2  = alloc((size_t)Bg * K1C * HID);
  float* s2   = alloc((size_t)Bg * K1C * K2C);
  float* As2  = alloc((size_t)Bg * K1C * K2C);
  float* x2   = alloc((size_t)Bg * K2C * HID);
  float* adj2 = alloc((size_t)Bg * K2C * K2C);
  float* ss2  = alloc((size_t)Bg * K2C * K2C);
  float* num2 = alloc(Bg); float* den2 = alloc(Bg); float* or2 = alloc(Bg);
  float* dv2  = alloc((size_t)Bg * K2C);
  float* t3   = alloc((size_t)Bg * K2C * HID);
  float* xd3  = alloc((size_t)Bg * K2C * HID);
  float* gm   = alloc((size_t)Bg * HID);
  float* g1   = alloc((size_t)Bg * HID);

  // zero scatter targets (adj, degin, rowsum are contiguous)
  hipMemsetAsync(adj, 0, ((size_t)Bg * Nn * Nn + 2 * (size_t)TOT) * sizeof(float), stream);

  k_build_adj<<<(E + 255) / 256, 256, 0, stream>>>(ei, E, adj, degin, rowsum);
  k_xw<<<dim3(TOT / 16, HID / 16), 32, 0, stream>>>(x, c1W, xw);
  k_gcn<<<Bg, 1024, (Nn * HID + Nn) * sizeof(float), stream>>>(adj, xw, degin, c1b, hfeat);
  {
    size_t tot = (size_t)TOT * K1C;
    k_s1lin<<<(int)((tot + 255) / 256), 256, 0, stream>>>(hfeat, p1W, p1b, s1);
  }
  k_softmax100<<<TOT, 32, 0, stream>>>(s1);
  k_pool_x<<<dim3(7, HID / 16, Bg), 32, 0, stream>>>(s1, hfeat, x1);
  k_As<<<Bg, 1024, Nn * K1C * sizeof(float), stream>>>(adj, s1, As1);
  k_sTB<<<dim3(7, 7, Bg), 32, 0, stream>>>(s1, As1, adj1);
  k_sTB<<<dim3(7, 7, Bg), 32, 0, stream>>>(s1, s1, ss1);
  k_losses<<<Bg, 256, 0, stream>>>(adj1, ss1, s1, rowsum, Nn, K1C, num1, den1, or1);
  k_adjd<<<Bg, 128, 0, stream>>>(adj1, dv1, K1C);
  {
    size_t tot = (size_t)Bg * K1C * K1C;
    k_adjnorm<<<(int)((tot + 255) / 256), 256, 0, stream>>>(adj1, dv1, K1C);
  }
  {
    size_t tot = (size_t)Bg * K1C;
    k_rowsum<<<(int)((tot + 255) / 256), 256, 0, stream>>>(adj1, dfl1, K1C);
  }
  {
    size_t tot = (size_t)Bg * K1C * HID;
    k_dmm_adj_x<<<(int)((tot + 255) / 256), 256, 0, stream>>>(adj1, x1, t2, K1C, HID);
    k_gconv_fin<<<(int)((tot + 255) / 256), 256, 0, stream>>>(t2, x1, c2rW, c2rb, c2oW, xd2, K1C, 1);
  }
  {
    size_t tot = (size_t)Bg * K1C * K2C;
    k_s2lin<<<(int)((tot + 255) / 256), 256, 0, stream>>>(xd2, p2W, p2b, s2);
  }
  {
    size_t tot = (size_t)Bg * K1C;
    k_softmax10<<<(int)((tot + 255) / 256), 256, 0, stream>>>(s2);
  }
  {
    size_t tot = (size_t)Bg * K2C * HID;
    k_pool_small<<<(int)((tot + 255) / 256), 256, 0, stream>>>(s2, xd2, x2, K1C, K2C, HID);
  }
  {
    size_t tot = (size_t)Bg * K1C * K2C;
    k_dmm_adj_x<<<(int)((tot + 255) / 256), 256, 0, stream>>>(adj1, s2, As2, K1C, K2C);
  }
  {
    size_t tot = (size_t)Bg * K2C * K2C;
    k_pool_small<<<(int)((tot + 255) / 256), 256, 0, stream>>>(s2, As2, adj2, K1C, K2C, K2C);
    k_pool_small<<<(int)((tot + 255) / 256), 256, 0, stream>>>(s2, s2, ss2, K1C, K2C, K2C);
  }
  k_losses<<<Bg, 256, 0, stream>>>(adj2, ss2, s2, dfl1, K1C, K2C, num2, den2, or2);
  k_adjd<<<Bg, 32, 0, stream>>>(adj2, dv2, K2C);
  {
    size_t tot = (size_t)Bg * K2C * K2C;
    k_adjnorm<<<(int)((tot + 255) / 256), 256, 0, stream>>>(adj2, dv2, K2C);
  }
  {
    size_t tot = (size_t)Bg * K2C * HID;
    k_dmm_adj_x<<<(int)((tot + 255) / 256), 256, 0, stream>>>(adj2, x2, t3, K2C, HID);
    k_gconv_fin<<<(int)((tot + 255) / 256), 256, 0, stream>>>(t3, x2, c3rW, c3rb, c3oW, xd3, K2C, 0);
  }
  {
    size_t tot = (size_t)Bg * HID;
    k_mean<<<(int)((tot + 255) / 256), 256, 0, stream>>>(xd3, gm);
    k_lin1<<<(int)((tot + 255) / 256), 256, 0, stream>>>(gm, l1W, l1b, g1);
  }
  k_head<<<1, 128, 0, stream>>>(g1, l2W, l2b, out);
  k_floss<<<1, 128, 0, stream>>>(num1, den1, or1, num2, den2, or2, out);
}